// SparseAttentionStrided_36575941493069
// MI455X (gfx1250) — compile-verified
//
#include <hip/hip_runtime.h>
#include <stdint.h>

// Sparse-attention strided/local mask generator for MI455X (gfx1250).
// Output: [1, 16, 4096, 4096] f32, 1 GiB. Pure store-bandwidth problem
// (reference ignores qk values). Strategy: build mask patterns once in LDS,
// then stream them to HBM with CDNA5 async LDS->global B128 stores (ASYNCcnt
// path), non-temporal hint since 1 GiB >> 192 MB L2 and is never re-read.

#define SEQ   4096
#define BAND  64          // l = floor(sqrt(4096))
#define NEGV  -10000.0f

#define TPL_LEN   8200                 // floats per shifted template copy
#define TPL_BYTES (TPL_LEN * 4)        // 32800 B, multiple of 16
#define LOCAL_LDS_BYTES (4 * TPL_BYTES)

__device__ __forceinline__ void async_store_b128(const float* base,
                                                 unsigned goff, unsigned lds) {
  // VADDR = per-lane 32-bit global byte offset, VSRC = LDS byte address,
  // SADDR = uniform 64-bit base. Tracked by ASYNCcnt.
  asm volatile("global_store_async_from_lds_b128 %0, %1, %2 th:TH_STORE_NT"
               :: "v"(goff), "v"(lds), "s"(base)
               : "memory");
}

// Heads 0..7: band mask. Each block: 8 rows x 8 heads = 1 MB of stores.
extern "C" __global__ void __launch_bounds__(256)
sas_local_kernel(float* __restrict__ out) {
  extern __shared__ float smem[];                 // 4 * TPL_LEN floats
  const unsigned tid = threadIdx.x;
  const unsigned i0  = blockIdx.x * 8u;

  // Build 4 shifted copies of the band template:
  //   T_p[x] = V[x - p],  V[d] = (4031 <= d <= 4159) ? 0 : -1e4
  for (unsigned idx = tid; idx < 4u * TPL_LEN; idx += 256u) {
    const int p = (int)(idx / TPL_LEN);
    const int x = (int)(idx % TPL_LEN);
    const int d = x - p;
    smem[idx] = (d >= (SEQ - 1 - BAND) && d <= (SEQ - 1 + BAND)) ? 0.0f : NEGV;
  }
  __syncthreads();

  for (int r = 0; r < 8; ++r) {
    const int i = (int)i0 + r;
    const int o = (SEQ - 1) - i;                  // row i = V[o .. o+4095]
    const int p = (4 - (o & 3)) & 3;              // pick copy so start is 16B-aligned
    const int s = o + p;                          // multiple of 4 (floats)
    const unsigned ldsRow = (unsigned)p * (unsigned)TPL_BYTES + (unsigned)s * 4u;
    for (int h = 0; h < 8; ++h) {                 // 8 identical local heads
      const float* rb = out + ((size_t)(h * SEQ + i)) * (size_t)SEQ;
#pragma unroll
      for (int k = 0; k < 4; ++k) {               // 256 thr * 16 B * 4 = 16 KB row
        const unsigned goff = tid * 16u + (unsigned)k * 4096u;
        async_store_b128(rb, goff, ldsRow + goff);
      }
    }
  }
  asm volatile("s_wait_asynccnt 0" ::: "memory");
}

// Heads 8..15: strided mask, rows periodic with 64 floats (256 B).
extern "C" __global__ void __launch_bounds__(256)
sas_strided_kernel(float* __restrict__ out) {
  extern __shared__ float smem[];                 // 8 * 64 floats = 2 KB
  const unsigned tid = threadIdx.x;
  const unsigned i0  = blockIdx.x * 8u;

  // Pattern for row class c = i%64: pat[jj] = (jj == c) ? 0 : -1e4
  for (unsigned idx = tid; idx < 512u; idx += 256u) {
    const unsigned r  = idx >> 6;
    const unsigned jj = idx & 63u;
    const unsigned c  = (i0 + r) & 63u;
    smem[idx] = (jj == c) ? 0.0f : NEGV;
  }
  __syncthreads();

  for (int r = 0; r < 8; ++r) {
    const int i = (int)i0 + r;
    // Each lane's 16 B chunk wraps inside the 256 B period:
    // global float offset f = tid*4 + k*1024; f % 64 == (tid&15)*4 for all k.
    const unsigned ldsRow = (unsigned)r * 256u + (tid & 15u) * 16u;
    for (int h = 8; h < 16; ++h) {
      const float* rb = out + ((size_t)(h * SEQ + i)) * (size_t)SEQ;
#pragma unroll
      for (int k = 0; k < 4; ++k) {
        const unsigned goff = tid * 16u + (unsigned)k * 4096u;
        async_store_b128(rb, goff, ldsRow);
      }
    }
  }
  asm volatile("s_wait_asynccnt 0" ::: "memory");
}

extern "C" void kernel_launch(void* const* d_in, const int* in_sizes, int n_in,
                              void* d_out, int out_size, void* d_ws, size_t ws_size,
                              hipStream_t stream) {
  (void)d_in; (void)in_sizes; (void)n_in; (void)d_ws; (void)ws_size; (void)out_size;
  float* out = (float*)d_out;

  // 128 KB dynamic LDS for the band templates (WGP has 320 KB).
  hipFuncSetAttribute(reinterpret_cast<const void*>(sas_local_kernel),
                      hipFuncAttributeMaxDynamicSharedMemorySize, LOCAL_LDS_BYTES);

  // 4096 rows / 8 rows-per-block = 512 blocks each; 1 MB of stores per block.
  sas_local_kernel<<<512, 256, LOCAL_LDS_BYTES, stream>>>(out);
  sas_strided_kernel<<<512, 256, 2048, stream>>>(out);
}